// FactorizedBilinearPooling_50508815401696
// MI455X (gfx1250) — compile-verified
//
#include <hip/hip_runtime.h>
#include <hip/hip_bf16.h>

// ---------------------------------------------------------------------------
// FactorizedBilinearPooling reduction for MI455X (gfx1250).
//
// Per (input, b, c) slice of 32x32x32 fp32:
//   S = sum_over_2x2x2_windows( max(window) ) + sum(all)/8
// pooled[b,c] = (Sx+Sy)^2 + (Sy+Sz)^2 + (Sx+Sz)^2 ; L2-normalize over c.
//
// HBM-bound: 402 MB read -> ~17.3 us floor at 23.3 TB/s. Each slice streams
// through a 4-deep LDS pipeline driven by the CDNA5 async data mover
// (global_load_async_to_lds_b128, s_wait_asynccnt, s_wait_dscnt).
//
// Key property: producer wave k fills LDS bytes [512k, 512k+512) of each
// plane, and consumer wave k's windows read exactly that range -> the
// streaming loop is wave-private and needs NO block barriers; each wave
// free-runs on its own async/ds counters.
// ---------------------------------------------------------------------------

#define SLICE_FLOATS  32768           // 32*32*32
#define CHUNK_BYTES   8192            // 2 h-planes (2 * 32*32 * 4B)
#define PLANE_BYTES   4096
#define NCHUNKS       16
#define NBUF          4               // pipeline depth: 3 chunks in flight
#define BUF_BYTES     CHUNK_BYTES
#define LDS_BYTES     (NBUF * BUF_BYTES)

__global__ __launch_bounds__(256) void fbp_pool_reduce(
    const float* __restrict__ x,
    const float* __restrict__ y,
    const float* __restrict__ z,
    float* __restrict__ ws)
{
    __shared__ __align__(16) unsigned char smem[LDS_BYTES];
    __shared__ float red[256];

    const int t   = threadIdx.x;       // 0..255
    const int gid = blockIdx.x;        // 0..3071
    const int inp = gid >> 10;         // 0..2  (x, y, z)
    const int bc  = gid & 1023;        // (b,c) flat

    const float* src = (inp == 0) ? x : ((inp == 1) ? y : z);
    const unsigned long long gslice =
        (unsigned long long)(uintptr_t)(src + (size_t)bc * SLICE_FLOATS);

    // Low 32 bits of a flat LDS pointer == byte offset in the wave's LDS.
    const unsigned lbase = (unsigned)(uintptr_t)(&smem[0]);
    const unsigned lthr  = lbase + (unsigned)(t * 16);          // producer LDS
    const unsigned long long gthr = gslice + (unsigned long long)(t * 16);

    // Two b128 async ops move one 8KB chunk (2 planes) per 256-thread block;
    // per wave that is 2 instructions covering its own 2x512B LDS region.
    auto issue = [&](int chunk, int buf) {
        unsigned l0 = lthr + (unsigned)(buf * BUF_BYTES);
        unsigned l1 = l0 + PLANE_BYTES;
        unsigned long long g0 = gthr + (unsigned long long)(chunk * CHUNK_BYTES);
        unsigned long long g1 = g0 + PLANE_BYTES;
        asm volatile("global_load_async_to_lds_b128 %0, %1, off"
                     :: "v"(l0), "v"(g0) : "memory");
        asm volatile("global_load_async_to_lds_b128 %0, %1, off"
                     :: "v"(l1), "v"(g1) : "memory");
    };

    // Prime: chunks 0..2 into buffers 0..2 (6 async ops in flight per wave).
    issue(0, 0);
    issue(1, 1);
    issue(2, 2);

    // Consumer mapping: one 2x2x2 window per thread per chunk.
    const int d2 = t & 15;             // pooled D index
    const int w2 = t >> 4;             // pooled W index
    const unsigned coff = (unsigned)(w2 * 256 + d2 * 8);  // bytes within buf

    float acc = 0.0f;

#pragma unroll
    for (int k = 0; k < NCHUNKS; ++k) {
        if (k + 3 < NCHUNKS) {
            // Buffer (k+3)&3 was last read at iteration k-1 by THIS wave;
            // async LDS writes are unordered with DS ops, so drain DS reads
            // before the data mover may overwrite that region.
            asm volatile("s_wait_dscnt 0x0" ::: "memory");
            issue(k + 3, (k + 3) & (NBUF - 1));
        }
        // Steady state: leave 3 chunks (6 ops) in flight; taper at the tail.
        const int rem = (NCHUNKS - 1 - k) * 2;
        const int wn  = rem > 6 ? 6 : rem;
        asm volatile("s_wait_asynccnt %0" :: "i"(wn) : "memory");

        const unsigned char* b = smem + (k & (NBUF - 1)) * BUF_BYTES + coff;
        const float2 v00 = *(const float2*)(b);                     // h0,w0
        const float2 v01 = *(const float2*)(b + 128);               // h0,w1
        const float2 v10 = *(const float2*)(b + PLANE_BYTES);       // h1,w0
        const float2 v11 = *(const float2*)(b + PLANE_BYTES + 128); // h1,w1

        const float m = fmaxf(fmaxf(fmaxf(v00.x, v00.y), fmaxf(v01.x, v01.y)),
                              fmaxf(fmaxf(v10.x, v10.y), fmaxf(v11.x, v11.y)));
        const float s = (v00.x + v00.y) + (v01.x + v01.y) +
                        (v10.x + v10.y) + (v11.x + v11.y);
        acc += m + s * 0.125f;
    }

    // Block reduction of the 256 partial sums (separate LDS region, so no
    // aliasing with still-running waves' stream buffers).
    red[t] = acc;
    __syncthreads();
    for (int sdx = 128; sdx > 0; sdx >>= 1) {
        if (t < sdx) red[t] += red[t + sdx];
        __syncthreads();
    }
    if (t == 0) ws[gid] = red[0];
}

// pooled + L2 normalize over the 256-channel dim. Grid = 4 (one per batch).
__global__ __launch_bounds__(256) void fbp_finalize(
    const float* __restrict__ ws, float* __restrict__ out)
{
    __shared__ float red[256];
    const int b  = blockIdx.x;
    const int c  = threadIdx.x;
    const int bc = b * 256 + c;

    const float sx = ws[bc];
    const float sy = ws[1024 + bc];
    const float sz = ws[2048 + bc];

    const float a = sx + sy;
    const float d = sy + sz;
    const float e = sx + sz;
    const float p = a * a + d * d + e * e;

    red[c] = p * p;
    __syncthreads();
    for (int s = 128; s > 0; s >>= 1) {
        if (c < s) red[c] += red[c + s];
        __syncthreads();
    }
    const float norm = sqrtf(red[0]);
    out[bc] = p / fmaxf(norm, 1e-12f);
}

extern "C" void kernel_launch(void* const* d_in, const int* in_sizes, int n_in,
                              void* d_out, int out_size, void* d_ws, size_t ws_size,
                              hipStream_t stream)
{
    const float* x = (const float*)d_in[0];
    const float* y = (const float*)d_in[1];
    const float* z = (const float*)d_in[2];
    float* ws  = (float*)d_ws;   // 3 * 1024 floats = 12 KiB
    float* out = (float*)d_out;  // 4 * 256 floats

    // One block per (input, b, c) slice: 3 * 4 * 256 = 3072 blocks.
    fbp_pool_reduce<<<3072, 256, 0, stream>>>(x, y, z, ws);
    fbp_finalize<<<4, 256, 0, stream>>>(ws, out);
}